// AttentivePooling_63050119905437
// MI455X (gfx1250) — compile-verified
//
#include <hip/hip_runtime.h>

#define N_B 8
#define N_S 4096
#define N_E 1024
#define N_Q 4
#define N_H 4
#define N_D 256
#define HQ  16
#define SCALE (1.0f / 16.0f)   // 1/sqrt(256)

typedef __attribute__((ext_vector_type(2))) float v2f;
typedef __attribute__((ext_vector_type(8))) float v8f;

// ---- workspace layout (float offsets) ----
#define OFF_QTMP   0         // [4][1024]
#define OFF_QEFF   4096      // [16][1024]
#define OFF_CTERM  20480     // [16]
#define OFF_POOLH  20512     // [8][16][1024]
#define OFF_CTX    151584    // [8][4][1024]
#define OFF_CTXAVG 184352    // [8][1024]
#define OFF_SCORES 192544    // [8][4096][16]  (scores, then attn in-place)

__device__ __forceinline__ float waveReduce(float x) {
#pragma unroll
    for (int off = 16; off > 0; off >>= 1)
        x += __shfl_down(x, off, 32);
    return x;
}

// q[qi][e] = queries[qi] . Wq[e] + bq[e]   (wave per output, coalesced)
__global__ void compute_q(const float* __restrict__ queries,
                          const float* __restrict__ w_in,
                          const float* __restrict__ b_in,
                          float* __restrict__ qtmp) {
    int gw   = (blockIdx.x * blockDim.x + threadIdx.x) >> 5;  // 0..4095
    int lane = threadIdx.x & 31;
    int qi = gw >> 10, e = gw & 1023;
    const float* qr = queries + (size_t)qi * N_E;
    const float* wr = w_in + (size_t)e * N_E;
    float acc = 0.f;
#pragma unroll 4
    for (int i = lane; i < N_E; i += 32) acc += qr[i] * wr[i];
    acc = waveReduce(acc);
    if (lane == 0) qtmp[gw] = acc + b_in[e];
}

// qeff[hq][e] = scale * sum_d q[qi][h*256+d] * Wk[h*256+d][e] ; cterm[hq]
__global__ void compute_qeff(const float* __restrict__ qtmp,
                             const float* __restrict__ w_in,
                             const float* __restrict__ b_in,
                             float* __restrict__ qeff,
                             float* __restrict__ cterm) {
    int t = blockIdx.x * 256 + threadIdx.x;   // 16384
    int hq = t >> 10, e = t & 1023;
    int h = hq >> 2, qi = hq & 3;
    const float* qr = qtmp + (size_t)qi * N_E + h * N_D;
    float acc = 0.f;
#pragma unroll 4
    for (int d = 0; d < N_D; ++d)
        acc += qr[d] * w_in[(size_t)(N_E + h * N_D + d) * N_E + e]; // coalesced in e
    qeff[t] = acc * SCALE;
    if (e == 0) {
        float cb = 0.f;
        for (int d = 0; d < N_D; ++d) cb += qr[d] * b_in[N_E + h * N_D + d];
        cterm[hq] = cb * SCALE;
    }
}

// scores[b][s][hq] = hidden[b,s,:] . qeff[hq,:] + cterm[hq]
// WMMA: M = 16 s-rows (A, from hidden), N = 16 hq (B, from LDS qeff), K = e
__global__ void scores_wmma(const float* __restrict__ hidden,
                            const float* __restrict__ qeff,
                            const float* __restrict__ cterm,
                            float* __restrict__ scores) {
    __shared__ float qs[HQ * N_E];   // 64KB, e-major: qs[e*16 + hq] -> conflict-free
    for (int i = threadIdx.x; i < HQ * N_E; i += blockDim.x) {
        int hq = i >> 10, e = i & 1023;
        qs[e * HQ + hq] = qeff[i];
    }
    __syncthreads();

    int lane = threadIdx.x & 31;
    int gw   = (blockIdx.x * blockDim.x + threadIdx.x) >> 5;  // 0..2047
    int b  = gw >> 8;
    int s0 = (gw & 255) << 4;
    int mrow = lane & 15;            // A: row M ; B: col N ; D: col N
    int g2   = (lane >> 4) << 1;     // hi half-wave handles K+2,K+3

    const float* aptr = hidden + ((size_t)b * N_S + (s0 + mrow)) * N_E + g2;
    v8f c = {};
#pragma unroll 4
    for (int k0 = 0; k0 < N_E; k0 += 4) {
        v2f a = *(const v2f*)(aptr + k0);                // hidden row, contiguous per lane
        v2f bm;
        bm.x = qs[(k0 + g2) * HQ + mrow];
        bm.y = qs[(k0 + g2 + 1) * HQ + mrow];
        c = __builtin_amdgcn_wmma_f32_16x16x4_f32(false, a, false, bm,
                                                  (short)0, c, false, false);
    }
    int   ghalf = lane >> 4;
    float ct = cterm[mrow];
    float* dst = scores + ((size_t)b * N_S + s0) * HQ + mrow;
#pragma unroll
    for (int j = 0; j < 8; ++j) {
        int m = j + ghalf * 8;                            // D: VGPR j -> M=j / j+8
        dst[(size_t)m * HQ] = c[j] + ct;
    }
}

// softmax over s for each (b,hq); in place
__global__ void softmax_kernel(float* __restrict__ scores) {
    __shared__ float red[256];
    int b = blockIdx.x >> 4, hq = blockIdx.x & 15;
    float* base = scores + (size_t)b * N_S * HQ + hq;
    int t = threadIdx.x;
    float v[16];
    float mx = -__builtin_inff();
#pragma unroll
    for (int i = 0; i < 16; ++i) {
        v[i] = base[(size_t)(t + i * 256) * HQ];
        mx = fmaxf(mx, v[i]);
    }
    red[t] = mx; __syncthreads();
    for (int off = 128; off > 0; off >>= 1) {
        if (t < off) red[t] = fmaxf(red[t], red[t + off]);
        __syncthreads();
    }
    mx = red[0]; __syncthreads();
    float sum = 0.f;
#pragma unroll
    for (int i = 0; i < 16; ++i) { v[i] = __expf(v[i] - mx); sum += v[i]; }
    red[t] = sum; __syncthreads();
    for (int off = 128; off > 0; off >>= 1) {
        if (t < off) red[t] += red[t + off];
        __syncthreads();
    }
    float inv = 1.f / red[0];
#pragma unroll
    for (int i = 0; i < 16; ++i) base[(size_t)(t + i * 256) * HQ] = v[i] * inv;
}

// attn_weights[b][q][s] = mean over heads
__global__ void attn_weights_kernel(const float* __restrict__ attn,
                                    float* __restrict__ aw) {
    int t = blockIdx.x * 256 + threadIdx.x;    // 131072
    int b = t >> 14, r = t & 16383;
    int qi = r >> 12, s = r & 4095;
    const float* p = attn + ((size_t)b * N_S + s) * HQ + qi;
    aw[t] = 0.25f * (p[0] + p[4] + p[8] + p[12]);
}

// poolH[b][hq][e] += sum_s attn[b][s][hq] * hidden[b][s][e]
// WMMA: M = hq(16) (A from attn), N = 16 e-cols (B from hidden, coalesced), K = s
__global__ void pool_wmma(const float* __restrict__ attn,
                          const float* __restrict__ hidden,
                          float* __restrict__ poolH) {
    int lane = threadIdx.x & 31;
    int gw   = (blockIdx.x * blockDim.x + threadIdx.x) >> 5;  // 0..2047
    int schunk = gw & 3;
    int etile  = (gw >> 2) & 63;
    int b      = gw >> 8;
    int e0 = etile << 4;
    int sb = schunk << 10;
    int nl = lane & 15;
    int g2 = (lane >> 4) << 1;
    const float* abase = attn   + (size_t)b * N_S * HQ  + nl;       // A row = hq
    const float* bbase = hidden + (size_t)b * N_S * N_E + e0 + nl;  // B col = e
    v8f c = {};
#pragma unroll 2
    for (int k0 = sb; k0 < sb + 1024; k0 += 4) {
        v2f a, bm;
        a.x  = abase[(size_t)(k0 + g2) * HQ];
        a.y  = abase[(size_t)(k0 + g2 + 1) * HQ];
        bm.x = bbase[(size_t)(k0 + g2) * N_E];
        bm.y = bbase[(size_t)(k0 + g2 + 1) * N_E];
        c = __builtin_amdgcn_wmma_f32_16x16x4_f32(false, a, false, bm,
                                                  (short)0, c, false, false);
    }
    int ghalf = lane >> 4;
    float* dst = poolH + (size_t)b * HQ * N_E + e0 + nl;
#pragma unroll
    for (int j = 0; j < 8; ++j) {
        int m = j + ghalf * 8;
        atomicAdd(dst + (size_t)m * N_E, c[j]);
    }
}

// ctx[b][qi][ep] = poolH[b][hq] . Wv[ep] + bv[ep]   (wave per output)
__global__ void ctx_kernel(const float* __restrict__ poolH,
                           const float* __restrict__ w_in,
                           const float* __restrict__ b_in,
                           float* __restrict__ ctx) {
    int gw   = (blockIdx.x * blockDim.x + threadIdx.x) >> 5;  // 0..32767
    int lane = threadIdx.x & 31;
    int b = gw >> 12, r = gw & 4095;
    int qi = r >> 10, ep = r & 1023;
    int hq = (ep >> 8) * 4 + qi;
    const float* prow = poolH + ((size_t)b * HQ + hq) * N_E;
    const float* wrow = w_in + (size_t)(2 * N_E + ep) * N_E;
    float acc = 0.f;
#pragma unroll 4
    for (int e = lane; e < N_E; e += 32) acc += prow[e] * wrow[e];
    acc = waveReduce(acc);
    if (lane == 0) ctx[gw] = acc + b_in[2 * N_E + ep];
}

__global__ void avg_ctx_kernel(const float* __restrict__ ctx,
                               float* __restrict__ ctxavg) {
    int t = blockIdx.x * 256 + threadIdx.x;   // 8192
    int b = t >> 10, ep = t & 1023;
    const float* p = ctx + (size_t)b * 4096 + ep;
    ctxavg[t] = 0.25f * (p[0] + p[1024] + p[2048] + p[3072]);
}

// pooled[b][e] = ctxavg[b] . out_w[e] + out_b[e]   (wave per output)
__global__ void out_kernel(const float* __restrict__ ctxavg,
                           const float* __restrict__ w_out,
                           const float* __restrict__ b_out,
                           float* __restrict__ out) {
    int gw   = (blockIdx.x * blockDim.x + threadIdx.x) >> 5;  // 0..8191
    int lane = threadIdx.x & 31;
    int b = gw >> 10, e = gw & 1023;
    const float* ca = ctxavg + (size_t)b * N_E;
    const float* wr = w_out + (size_t)e * N_E;
    float acc = 0.f;
#pragma unroll 4
    for (int ep = lane; ep < N_E; ep += 32) acc += ca[ep] * wr[ep];
    acc = waveReduce(acc);
    if (lane == 0) out[gw] = acc + b_out[e];
}

extern "C" void kernel_launch(void* const* d_in, const int* in_sizes, int n_in,
                              void* d_out, int out_size, void* d_ws, size_t ws_size,
                              hipStream_t stream) {
    const float* hidden  = (const float*)d_in[0];
    const float* queries = (const float*)d_in[1];
    const float* w_in    = (const float*)d_in[2];
    const float* b_in    = (const float*)d_in[3];
    const float* w_out   = (const float*)d_in[4];
    const float* b_out   = (const float*)d_in[5];
    float* out = (float*)d_out;
    float* ws  = (float*)d_ws;

    hipMemsetAsync(ws + OFF_POOLH, 0, (size_t)N_B * HQ * N_E * sizeof(float), stream);

    compute_q<<<512, 256, 0, stream>>>(queries, w_in, b_in, ws + OFF_QTMP);
    compute_qeff<<<64, 256, 0, stream>>>(ws + OFF_QTMP, w_in, b_in,
                                         ws + OFF_QEFF, ws + OFF_CTERM);
    scores_wmma<<<256, 256, 0, stream>>>(hidden, ws + OFF_QEFF, ws + OFF_CTERM,
                                         ws + OFF_SCORES);
    softmax_kernel<<<128, 256, 0, stream>>>(ws + OFF_SCORES);
    attn_weights_kernel<<<512, 256, 0, stream>>>(ws + OFF_SCORES, out + N_B * N_E);
    pool_wmma<<<256, 256, 0, stream>>>(ws + OFF_SCORES, hidden, ws + OFF_POOLH);
    ctx_kernel<<<4096, 256, 0, stream>>>(ws + OFF_POOLH, w_in, b_in, ws + OFF_CTX);
    avg_ctx_kernel<<<32, 256, 0, stream>>>(ws + OFF_CTX, ws + OFF_CTXAVG);
    out_kernel<<<1024, 256, 0, stream>>>(ws + OFF_CTXAVG, w_out, b_out, out);
}